// bssubgnn_9311489098067
// MI455X (gfx1250) — compile-verified
//
#include <hip/hip_runtime.h>
#include <hip/hip_bf16.h>

typedef float v2f __attribute__((ext_vector_type(2)));
typedef float v8f __attribute__((ext_vector_type(8)));

// ---------------------------------------------------------------------------
// f32 WMMA GEMM: C[M,NTILE] = act(A[M,K] @ B[K,NTILE] + bias)
// One wave per 16-row strip; uses V_WMMA_F32_16X16X4_F32.
// A-fragment layout (ISA 7.12.2): lanes 0-15 -> K={k,k+1}, lanes 16-31 -> K={k+2,k+3}
// C/D layout: VGPR v, lanes 0-15: row m0+v; lanes 16-31: row m0+v+8; col = n0+(lane&15)
// ACT: 0=none, 1=tanh, 2=relu.  M must be a multiple of 16 (EXEC all-ones).
// ---------------------------------------------------------------------------
template <int NTILE, int ACT>
__global__ __launch_bounds__(256) void gemm_f32_wmma(
    const float* __restrict__ A, int lda,
    const float* __restrict__ B,           // [K, NTILE] row-major
    const float* __restrict__ bias,        // [NTILE] or nullptr
    float* __restrict__ C, int ldc,
    int M, int K)
{
    constexpr int NS = NTILE / 16;
    const int wave = threadIdx.x >> 5;
    const int lane = threadIdx.x & 31;
    const int m0 = (blockIdx.x * 8 + wave) * 16;
    if (m0 >= M) return;                    // wave-uniform exit: EXEC stays all-ones
    const int half = lane >> 4;
    const int l15  = lane & 15;

    v8f acc[NS] = {};
    const float* arow = A + (size_t)(m0 + l15) * lda + half * 2;

    for (int kk = 0; kk < K; kk += 4) {
        v2f a = *(const v2f*)(arow + kk);   // 8B vector load, contiguous K pair
        const float* bp = B + (size_t)(kk + half * 2) * NTILE + l15;
#pragma unroll
        for (int s = 0; s < NS; ++s) {
            v2f b;
            b.x = bp[s * 16];               // row kk+2*half
            b.y = bp[NTILE + s * 16];       // row kk+2*half+1
            acc[s] = __builtin_amdgcn_wmma_f32_16x16x4_f32(
                false, a, false, b, (short)0, acc[s], false, false);
        }
    }

#pragma unroll
    for (int s = 0; s < NS; ++s) {
        const float bv = bias ? bias[s * 16 + l15] : 0.0f;
#pragma unroll
        for (int v = 0; v < 8; ++v) {
            float val = acc[s][v] + bv;
            if (ACT == 1) val = tanhf(val);
            if (ACT == 2) val = fmaxf(val, 0.0f);
            C[(size_t)(m0 + v + half * 8) * ldc + s * 16 + l15] = val;
        }
    }
}

// ---------------------------------------------------------------------------
// Small helpers
// ---------------------------------------------------------------------------
__global__ void fill_f32(float* __restrict__ p, float v, long n) {
    long i = (long)blockIdx.x * blockDim.x + threadIdx.x;
    if (i < n) p[i] = v;
}

__global__ void deg_accum(const int* __restrict__ col, float* __restrict__ deg, int e) {
    int i = blockIdx.x * blockDim.x + threadIdx.x;
    if (i < e) atomicAdd(&deg[col[i]], 1.0f);
}

__global__ void rsqrt_k(float* __restrict__ d, int n) {
    int i = blockIdx.x * blockDim.x + threadIdx.x;
    if (i < n) d[i] = rsqrtf(d[i]);
}

// agg self-loop + bias init: hcat[i][128+j] = t[i][j]*dis[i]^2 + b_agg[j]
__global__ void agg_init(const float* __restrict__ t, const float* __restrict__ dis,
                         const float* __restrict__ bagg, float* __restrict__ hcat, int n) {
    int idx = blockIdx.x * blockDim.x + threadIdx.x;
    if (idx >= n * 128) return;
    int i = idx >> 7, j = idx & 127;
    float di = dis[i];
    hcat[(size_t)i * 256 + 128 + j] = t[(size_t)i * 128 + j] * di * di + bagg[j];
}

// one wave per edge: hcat[c][128+*] += t[r][*] * dis[r]*dis[c]
__global__ __launch_bounds__(256) void scatter_edges(
    const float* __restrict__ t, const float* __restrict__ dis,
    const int* __restrict__ row, const int* __restrict__ col,
    float* __restrict__ hcat, int e)
{
    int wid = (int)(((long)blockIdx.x * blockDim.x + threadIdx.x) >> 5);
    if (wid >= e) return;
    int lane = threadIdx.x & 31;
    int r = row[wid], c = col[wid];
    float nm = dis[r] * dis[c];
    float4 v = ((const float4*)(t + (size_t)r * 128))[lane];  // coalesced 512B/wave
    float* dst = hcat + (size_t)c * 256 + 128 + lane * 4;
    atomicAdd(dst + 0, v.x * nm);
    atomicAdd(dst + 1, v.y * nm);
    atomicAdd(dst + 2, v.z * nm);
    atomicAdd(dst + 3, v.w * nm);
}

// one wave per node: scores[i][m] = exp(tanh(z_i . wpool_m + b_m)) * mask_m(i)
__global__ __launch_bounds__(128) void score_kernel(
    const float* __restrict__ x, const float* __restrict__ z,
    const float* __restrict__ w0, const float* __restrict__ b0,
    const float* __restrict__ w1, const float* __restrict__ b1,
    const float* __restrict__ w2, const float* __restrict__ b2,
    float* __restrict__ scores, int n)
{
    int wid = (int)(((long)blockIdx.x * blockDim.x + threadIdx.x) >> 5);
    if (wid >= n) return;
    int lane = threadIdx.x & 31;
    const float* zr = z + (size_t)wid * 384;
    float a0 = 0.f, a1 = 0.f, a2 = 0.f;
    for (int k = lane; k < 384; k += 32) {
        float zv = zr[k];
        a0 += zv * w0[k]; a1 += zv * w1[k]; a2 += zv * w2[k];
    }
    for (int off = 16; off; off >>= 1) {
        a0 += __shfl_down(a0, off);
        a1 += __shfl_down(a1, off);
        a2 += __shfl_down(a2, off);
    }
    if (lane == 0) {
        const float* xr = x + (size_t)wid * 32;
        float m0 = (xr[2] == 1.0f) ? 1.0f : 0.0f;
        float m1 = (xr[3] == 1.0f) ? 1.0f : 0.0f;
        float m2 = (xr[4] == 1.0f) ? 1.0f : 0.0f;
        scores[wid * 3 + 0] = expf(tanhf(a0 + b0[0])) * m0;
        scores[wid * 3 + 1] = expf(tanhf(a1 + b1[0])) * m1;
        scores[wid * 3 + 2] = expf(tanhf(a2 + b2[0])) * m2;
    }
}

__device__ __forceinline__ int lbound(const int* __restrict__ a, int n, int v) {
    int lo = 0, hi = n;
    while (lo < hi) { int mid = (lo + hi) >> 1; if (a[mid] < v) lo = mid + 1; else hi = mid; }
    return lo;
}

// block per subgraph (batch is sorted): no atomics
__global__ __launch_bounds__(384) void pool_kernel(
    const float* __restrict__ z, const float* __restrict__ scores,
    const int* __restrict__ batch, float* __restrict__ pooled, int n_nodes)
{
    __shared__ int s_lo, s_hi;
    int s = blockIdx.x;
    if (threadIdx.x == 0) {
        s_lo = lbound(batch, n_nodes, s);
        s_hi = lbound(batch, n_nodes, s + 1);
    }
    __syncthreads();
    int j = threadIdx.x;
    float a0 = 0.f, a1 = 0.f, a2 = 0.f;
    for (int i = s_lo; i < s_hi; ++i) {
        float zv = z[(size_t)i * 384 + j];
        a0 += zv * scores[i * 3 + 0];
        a1 += zv * scores[i * 3 + 1];
        a2 += zv * scores[i * 3 + 2];
    }
    float* pr = pooled + (size_t)s * 1152;
    pr[j] = a0; pr[384 + j] = a1; pr[768 + j] = a2;
}

// tiny tail MLP: relu(h@W1+b1) -> relu(@W2+b2) -> @W3+b3
__global__ void mlp_tail(const float* __restrict__ he0,
                         const float* __restrict__ w1, const float* __restrict__ b1,
                         const float* __restrict__ w2, const float* __restrict__ b2,
                         const float* __restrict__ w3, const float* __restrict__ b3,
                         float* __restrict__ out, int n_sub)
{
    int r = blockIdx.x * blockDim.x + threadIdx.x;
    if (r >= n_sub) return;
    float h1[32], h2[32], h3[16];
    const float* hr = he0 + (size_t)r * 32;
#pragma unroll
    for (int k = 0; k < 32; ++k) h1[k] = hr[k];
    for (int o = 0; o < 32; ++o) {
        float acc = b1[o];
#pragma unroll
        for (int k = 0; k < 32; ++k) acc += h1[k] * w1[k * 32 + o];
        h2[o] = fmaxf(acc, 0.0f);
    }
    for (int o = 0; o < 16; ++o) {
        float acc = b2[o];
#pragma unroll
        for (int k = 0; k < 32; ++k) acc += h2[k] * w2[k * 16 + o];
        h3[o] = fmaxf(acc, 0.0f);
    }
    for (int o = 0; o < 4; ++o) {
        float acc = b3[o];
#pragma unroll
        for (int k = 0; k < 16; ++k) acc += h3[k] * w3[k * 4 + o];
        out[(size_t)r * 4 + o] = acc;
    }
}

// ---------------------------------------------------------------------------
// Host orchestration
// ---------------------------------------------------------------------------
extern "C" void kernel_launch(void* const* d_in, const int* in_sizes, int n_in,
                              void* d_out, int out_size, void* d_ws, size_t ws_size,
                              hipStream_t stream)
{
    (void)n_in; (void)out_size; (void)ws_size;
    // Input order: x, 32 param leaves (insertion order), edge_index, batch, num_subgraphs
    const float* x = (const float*)d_in[0];
    auto P = [&](int i) { return (const float*)d_in[1 + i]; };
    const int* eidx  = (const int*)d_in[33];
    const int* batch = (const int*)d_in[34];

    const int N = in_sizes[0] / 32;        // 100000 (multiple of 16)
    const int E = in_sizes[33] / 2;        // 1600000
    const int NSUB = 1000;
    const int MP = 1024;                    // padded pooled rows for WMMA strips
    const int* row = eidx;
    const int* col = eidx + E;

    // workspace layout
    char* ws = (char*)d_ws;
    size_t off = 0;
    auto walloc = [&](size_t bytes) { void* p = ws + off; off = (off + bytes + 255) & ~(size_t)255; return p; };
    float* dis    = (float*)walloc((size_t)N * 4);
    float* hcat   = (float*)walloc((size_t)N * 256 * 4);   // [h1 | agg]
    float* tbuf   = (float*)walloc((size_t)N * 128 * 4);   // h1 @ W_agg
    float* z      = (float*)walloc((size_t)N * 384 * 4);   // layer outputs concat
    float* scores = (float*)walloc((size_t)N * 3 * 4);
    float* pooled = (float*)walloc((size_t)MP * 1152 * 4);
    float* he0    = (float*)walloc((size_t)MP * 32 * 4);

    // symmetric-norm degrees (self-loops folded into init value 1.0)
    fill_f32<<<(N + 255) / 256, 256, 0, stream>>>(dis, 1.0f, N);
    deg_accum<<<(E + 255) / 256, 256, 0, stream>>>(col, dis, E);
    rsqrt_k<<<(N + 255) / 256, 256, 0, stream>>>(dis, N);

    const int strips = N / 16;
    dim3 gblocks((strips + 7) / 8);

    for (int i = 0; i < 3; ++i) {
        const float* Ain = (i == 0) ? x : (z + (size_t)(i - 1) * 128);
        const int lda = (i == 0) ? 32 : 384;
        const int Kin = (i == 0) ? 32 : 128;
        const float* wlin = P(6 * i + 0); const float* blin = P(6 * i + 1);
        const float* wagg = P(6 * i + 2); const float* bagg = P(6 * i + 3);
        const float* wcat = P(6 * i + 4); const float* bcat = P(6 * i + 5);

        // h1 = A @ W_lin + b_lin  -> hcat[:, 0:128]
        gemm_f32_wmma<128, 0><<<gblocks, 256, 0, stream>>>(Ain, lda, wlin, blin, hcat, 256, N, Kin);
        // t = h1 @ W_agg          -> tbuf
        gemm_f32_wmma<128, 0><<<gblocks, 256, 0, stream>>>(hcat, 256, wagg, nullptr, tbuf, 128, N, 128);
        // agg init (self-loop + bias) -> hcat[:, 128:256]
        agg_init<<<(N * 128 + 255) / 256, 256, 0, stream>>>(tbuf, dis, bagg, hcat, N);
        // edge scatter-add
        scatter_edges<<<(E + 7) / 8, 256, 0, stream>>>(tbuf, dis, row, col, hcat, E);
        // h = tanh([h1|agg] @ W_cat + b_cat) -> z[:, i*128:(i+1)*128]
        gemm_f32_wmma<128, 1><<<gblocks, 256, 0, stream>>>(hcat, 256, wcat, bcat,
                                                           z + (size_t)i * 128, 384, N, 256);
    }

    // attention scores per mask
    score_kernel<<<(N + 3) / 4, 128, 0, stream>>>(x, z,
        P(18), P(19), P(20), P(21), P(22), P(23), scores, N);

    // masked pooled sums per subgraph (pad rows zeroed every call)
    fill_f32<<<((long)MP * 1152 + 255) / 256, 256, 0, stream>>>(pooled, 0.0f, (long)MP * 1152);
    pool_kernel<<<NSUB, 384, 0, stream>>>(z, scores, batch, pooled, N);

    // e0: relu(pooled @ W_e0 + b_e0), M padded to 1024
    gemm_f32_wmma<32, 2><<<(MP / 16 + 7) / 8, 256, 0, stream>>>(pooled, 1152, P(24), P(25), he0, 32, MP, 1152);

    // tail MLP -> d_out [1000,4]
    mlp_tail<<<(NSUB + 255) / 256, 256, 0, stream>>>(he0,
        P(26), P(27), P(28), P(29), P(30), P(31), (float*)d_out, NSUB);
}